// MultiHeadAttention_21414706937886
// MI455X (gfx1250) — compile-verified
//
#include <hip/hip_runtime.h>
#include <hip/hip_bf16.h>

// ---------------------------------------------------------------------------
// Causal MHA with T5 relative bias, fp32 end-to-end on CDNA5 WMMA.
// B=4, T=2048, C=1024, H=16, D=64, NUM_BUCKETS=33, scale = 1/8.
//
// Pipeline:
//   1) qkv_proj   : Q/K/V = X * W[h]     (WMMA f32 16x16x4, async double-
//                                         buffered LDS-tiled GEMM)
//   2) flash_attn : online-softmax attention, 16-row query tiles per wave
//   3) out_proj   : out = attnO * Wp + bp (same WMMA GEMM)
//
// Workspace: Q,K,V as (B,H,T,D) + attnO as (B,T,H*D), 4 * 32MB fp32 = 128MB.
// ---------------------------------------------------------------------------

typedef __attribute__((ext_vector_type(2))) float v2f;
typedef __attribute__((ext_vector_type(8))) float v8f;
typedef int v4i __attribute__((vector_size(16)));          // b128 payload
typedef __attribute__((address_space(1))) v4i gv4i;        // global int4
typedef __attribute__((address_space(3))) v4i lv4i;        // LDS int4

#define WMMA_F32(a, b, c) \
  __builtin_amdgcn_wmma_f32_16x16x4_f32(false, (a), false, (b), (short)0, (c), false, false)

#define MHA_B 4
#define MHA_T 2048
#define MHA_C 1024
#define MHA_H 16
#define MHA_D 64

// A staged 128x16 with row stride 20 floats (rows 0..15 map to distinct
// banks; row starts 80B => 16B aligned for b128 DMA writes).
#define ALDS_STRIDE 20
// B staged 16x64 with row stride 72 floats (288B, 16B aligned; rows k and k+2
// land 16 banks apart so the two half-waves hit disjoint banks).
#define BLDS_STRIDE 72

#define A_SLAB (128 * ALDS_STRIDE)
#define B_SLAB (16 * BLDS_STRIDE)

// ---------------------------------------------------------------------------
// Issue one 128x16 A-slab + 16x64 B-slab as async DMA into LDS.
// 3 async b128 ops per thread (2 for A, 1 for B), tracked by ASYNCcnt.
// ---------------------------------------------------------------------------
__device__ __forceinline__ void async_stage_slab(
    const float* __restrict__ A, int lda, const float* __restrict__ Bm,
    int ldb, int k0, float* a_buf, float* b_buf, int tid) {
#pragma unroll
  for (int i = 0; i < 2; ++i) {
    int f   = tid + i * 256;
    int row = f >> 2;
    int c4  = (f & 3) * 4;
    __builtin_amdgcn_global_load_async_to_lds_b128(
        (gv4i*)(A + (size_t)row * lda + k0 + c4),
        (lv4i*)(a_buf + row * ALDS_STRIDE + c4), 0, 0);
  }
  {
    int row = tid >> 4;
    int c4  = (tid & 15) * 4;
    __builtin_amdgcn_global_load_async_to_lds_b128(
        (gv4i*)(Bm + (size_t)(k0 + row) * ldb + c4),
        (lv4i*)(b_buf + row * BLDS_STRIDE + c4), 0, 0);
  }
}

// ---------------------------------------------------------------------------
// Shared 128(M) x 64(N) output-tile GEMM body. 256 threads = 8 waves (4Mx2N),
// each wave computes a 2x2 grid of 16x16 f32 WMMA fragments. K consumed in
// double-buffered async-DMA slabs of 16.
// ---------------------------------------------------------------------------
__device__ __forceinline__ void gemm_tile_128x64(
    const float* __restrict__ A, int lda,   // points at row mBase, col 0
    const float* __restrict__ Bm, int ldb,  // points at row 0, col nBase
    int K, float* a_lds, float* b_lds, v8f* c) {
  const int tid  = threadIdx.x;
  const int lane = tid & 31;
  const int wave = tid >> 5;
  const int wm   = (wave & 3) * 32;
  const int wn   = (wave >> 2) * 32;
  const int lm   = lane & 15;
  const int hi   = lane >> 4;
  const int nslab = K >> 4;

  async_stage_slab(A, lda, Bm, ldb, 0, a_lds, b_lds, tid);

  for (int i = 0; i < nslab; ++i) {
    float* a_cur = a_lds + (i & 1) * A_SLAB;
    float* b_cur = b_lds + (i & 1) * B_SLAB;
    if (i + 1 < nslab) {
      async_stage_slab(A, lda, Bm, ldb, (i + 1) * 16,
                       a_lds + ((i + 1) & 1) * A_SLAB,
                       b_lds + ((i + 1) & 1) * B_SLAB, tid);
      __builtin_amdgcn_s_wait_asynccnt(3);  // slab i landed (3 newer in flight)
    } else {
      __builtin_amdgcn_s_wait_asynccnt(0);
    }
    __syncthreads();  // all waves' DMA visible

#pragma unroll
    for (int kk = 0; kk < 16; kk += 4) {
      v2f a0, a1, b0, b1;
      const float* ap0 = a_cur + (wm + lm) * ALDS_STRIDE + kk + 2 * hi;
      const float* ap1 = ap0 + 16 * ALDS_STRIDE;
      a0.x = ap0[0]; a0.y = ap0[1];
      a1.x = ap1[0]; a1.y = ap1[1];
      const float* bp = b_cur + (kk + 2 * hi) * BLDS_STRIDE;
      b0.x = bp[wn + lm];      b0.y = bp[BLDS_STRIDE + wn + lm];
      b1.x = bp[wn + 16 + lm]; b1.y = bp[BLDS_STRIDE + wn + 16 + lm];
      c[0] = WMMA_F32(a0, b0, c[0]);
      c[1] = WMMA_F32(a0, b1, c[1]);
      c[2] = WMMA_F32(a1, b0, c[2]);
      c[3] = WMMA_F32(a1, b1, c[3]);
    }
    __syncthreads();  // done reading buf before it is DMA-overwritten
  }
}

// ---------------------------------------------------------------------------
// Kernel 1: per-head Q/K/V projections. grid = (M/128, H, 3).
// Output layout (B, H, T, D) for friendly attention access.
// ---------------------------------------------------------------------------
__global__ __launch_bounds__(256) void mha_qkv_proj(
    const float* __restrict__ x,
    const float* __restrict__ Wq, const float* __restrict__ Wk,
    const float* __restrict__ Wv,
    float* __restrict__ q, float* __restrict__ k, float* __restrict__ v) {
  __shared__ float a_lds[2 * A_SLAB];
  __shared__ float b_lds[2 * B_SLAB];

  const int h     = blockIdx.y;
  const int which = blockIdx.z;
  const float* W = (which == 0) ? Wq : (which == 1) ? Wk : Wv;
  float* out     = (which == 0) ? q  : (which == 1) ? k  : v;
  const int mBase = blockIdx.x * 128;

  v8f c[4] = {v8f{}, v8f{}, v8f{}, v8f{}};
  gemm_tile_128x64(x + (size_t)mBase * MHA_C, MHA_C,
                   W + (size_t)h * MHA_C * MHA_D, MHA_D,
                   MHA_C, a_lds, b_lds, c);

  const int tid = threadIdx.x, lane = tid & 31, wave = tid >> 5;
  const int wm = (wave & 3) * 32, wn = (wave >> 2) * 32;
  const int lm = lane & 15, hi = lane >> 4;
#pragma unroll
  for (int ti = 0; ti < 2; ++ti)
#pragma unroll
    for (int tj = 0; tj < 2; ++tj) {
      const v8f cc = c[ti * 2 + tj];
#pragma unroll
      for (int r = 0; r < 8; ++r) {
        int gm = mBase + wm + ti * 16 + r + 8 * hi;  // b*T + t
        int d  = wn + tj * 16 + lm;
        int bb = gm >> 11;          // / T
        int t  = gm & (MHA_T - 1);  // % T
        out[((((size_t)bb * MHA_H + h) * MHA_T + t) * MHA_D) + d] = cc[r];
      }
    }
}

// ---------------------------------------------------------------------------
// Kernel 2: flash attention with causal mask + T5 relative bias.
// grid = (T/128, H, B); 8 waves, each owns a 16-row query tile.
// ---------------------------------------------------------------------------
__global__ __launch_bounds__(256) void mha_flash_attn(
    const float* __restrict__ Q, const float* __restrict__ K,
    const float* __restrict__ V, const float* __restrict__ rel_table,
    float* __restrict__ O) {
  __shared__ float lds_rel[64];
  __shared__ float lds_p[8][16 * 17];  // per-wave P transpose buffer

  const int tid  = threadIdx.x;
  const int lane = tid & 31;
  const int wave = tid >> 5;
  const int lm   = lane & 15;
  const int hi   = lane >> 4;
  const int h    = blockIdx.y;
  const int bb   = blockIdx.z;
  const int q0   = blockIdx.x * 128 + wave * 16;

  if (tid < 33) lds_rel[tid] = rel_table[tid * MHA_H + h];  // (33,H) column h
  __syncthreads();

  const size_t bh = ((size_t)bb * MHA_H + h) * MHA_T;  // row base in (B*H, T, D)

  // Q fragments, pre-scaled by 1/sqrt(D) = 0.125
  v2f qa[16];
  const float* Qb = Q + (bh + q0) * MHA_D;
#pragma unroll
  for (int dk = 0; dk < 16; ++dk) {
    const float* p = Qb + (size_t)lm * MHA_D + dk * 4 + 2 * hi;
    qa[dk].x = p[0] * 0.125f;
    qa[dk].y = p[1] * 0.125f;
  }

  v8f o0 = {}, o1 = {}, o2 = {}, o3 = {};
  float mrow[8], lrow[8];
#pragma unroll
  for (int r = 0; r < 8; ++r) { mrow[r] = -3.0e38f; lrow[r] = 0.0f; }

  float* pw = &lds_p[wave][0];
  const int ntiles = (q0 >> 4) + 1;  // causal extent in 16-key tiles

  for (int kt = 0; kt < ntiles; ++kt) {
    const int s0 = kt * 16;
    const float* Kb = K + (bh + s0) * MHA_D;
    const float* Vb = V + (bh + s0) * MHA_D;
    if (kt + 1 < ntiles) {  // prefetch next key tile -> global_prefetch_b8
      __builtin_prefetch(Kb + (16 + lm) * MHA_D, 0, 3);
      __builtin_prefetch(Vb + (16 + lm) * MHA_D, 0, 3);
    }

    // ---- S = (Q*scale) K^T : 16 WMMA over D=64 ----
    v8f s = {};
#pragma unroll
    for (int dk = 0; dk < 16; ++dk) {
      v2f bf;
      const float* p = Kb + (size_t)lm * MHA_D + dk * 4 + 2 * hi;
      bf.x = p[0]; bf.y = p[1];
      s = WMMA_F32(qa[dk], bf, s);
    }

    // ---- bias + causal mask + online softmax (row groups of 16 lanes) ----
#pragma unroll
    for (int r = 0; r < 8; ++r) {
      const int t  = q0 + r + 8 * hi;
      const int ss = s0 + lm;
      int idx = ss - t + 16;
      idx = idx < 0 ? 0 : (idx > 32 ? 32 : idx);
      float val = (ss <= t) ? (s[r] + lds_rel[idx]) : -3.0e38f;

      float rm = val;
#pragma unroll
      for (int off = 8; off > 0; off >>= 1)
        rm = fmaxf(rm, __shfl_xor(rm, off, 16));
      const float mnew = fmaxf(mrow[r], rm);
      const float pv   = __expf(val - mnew);
      float rs = pv;
#pragma unroll
      for (int off = 8; off > 0; off >>= 1)
        rs += __shfl_xor(rs, off, 16);
      const float alpha = __expf(mrow[r] - mnew);
      lrow[r] = lrow[r] * alpha + rs;
      mrow[r] = mnew;
      o0[r] *= alpha; o1[r] *= alpha; o2[r] *= alpha; o3[r] *= alpha;

      pw[(r + 8 * hi) * 17 + lm] = pv;  // C-layout -> LDS for A-layout reload
    }
    asm volatile("s_wait_dscnt 0" ::: "memory");

    // ---- O += P V : 16 WMMA (4 k-steps x 4 d-tiles) ----
#pragma unroll
    for (int ks = 0; ks < 4; ++ks) {
      v2f ap;
      const float* pp = pw + lm * 17 + ks * 4 + 2 * hi;
      ap.x = pp[0]; ap.y = pp[1];
      const float* vp = Vb + (size_t)(ks * 4 + 2 * hi) * MHA_D;
      v2f b0, b1, b2, b3;
      b0.x = vp[lm];      b0.y = vp[MHA_D + lm];
      b1.x = vp[16 + lm]; b1.y = vp[MHA_D + 16 + lm];
      b2.x = vp[32 + lm]; b2.y = vp[MHA_D + 32 + lm];
      b3.x = vp[48 + lm]; b3.y = vp[MHA_D + 48 + lm];
      o0 = WMMA_F32(ap, b0, o0);
      o1 = WMMA_F32(ap, b1, o1);
      o2 = WMMA_F32(ap, b2, o2);
      o3 = WMMA_F32(ap, b3, o3);
    }
  }

  // ---- normalize and store in (B, T, H*D) layout ----
#pragma unroll
  for (int r = 0; r < 8; ++r) {
    const float inv = 1.0f / lrow[r];
    const int t = q0 + r + 8 * hi;
    float* op = O + ((size_t)bb * MHA_T + t) * (MHA_H * MHA_D) + h * MHA_D;
    op[lm]      = o0[r] * inv;
    op[16 + lm] = o1[r] * inv;
    op[32 + lm] = o2[r] * inv;
    op[48 + lm] = o3[r] * inv;
  }
}

// ---------------------------------------------------------------------------
// Kernel 3: output projection (8192x1024) x (1024x1024) + bias.
// grid = (M/128, N/64).
// ---------------------------------------------------------------------------
__global__ __launch_bounds__(256) void mha_out_proj(
    const float* __restrict__ A, const float* __restrict__ Wp,
    const float* __restrict__ bp, float* __restrict__ out) {
  __shared__ float a_lds[2 * A_SLAB];
  __shared__ float b_lds[2 * B_SLAB];

  const int mBase = blockIdx.x * 128;
  const int nBase = blockIdx.y * 64;

  v8f c[4] = {v8f{}, v8f{}, v8f{}, v8f{}};
  gemm_tile_128x64(A + (size_t)mBase * (MHA_H * MHA_D), MHA_H * MHA_D,
                   Wp + nBase, MHA_C, MHA_H * MHA_D, a_lds, b_lds, c);

  const int tid = threadIdx.x, lane = tid & 31, wave = tid >> 5;
  const int wm = (wave & 3) * 32, wn = (wave >> 2) * 32;
  const int lm = lane & 15, hi = lane >> 4;
#pragma unroll
  for (int ti = 0; ti < 2; ++ti)
#pragma unroll
    for (int tj = 0; tj < 2; ++tj) {
      const v8f cc = c[ti * 2 + tj];
#pragma unroll
      for (int r = 0; r < 8; ++r) {
        int gm = mBase + wm + ti * 16 + r + 8 * hi;
        int gn = nBase + wn + tj * 16 + lm;
        out[(size_t)gm * MHA_C + gn] = cc[r] + bp[gn];
      }
    }
}

// ---------------------------------------------------------------------------
extern "C" void kernel_launch(void* const* d_in, const int* in_sizes, int n_in,
                              void* d_out, int out_size, void* d_ws, size_t ws_size,
                              hipStream_t stream) {
  const float* x   = (const float*)d_in[0];
  const float* Wq  = (const float*)d_in[1];
  const float* Wk  = (const float*)d_in[2];
  const float* Wv  = (const float*)d_in[3];
  const float* Wp  = (const float*)d_in[4];
  const float* bp  = (const float*)d_in[5];
  const float* rel = (const float*)d_in[6];
  float* out = (float*)d_out;

  float* ws = (float*)d_ws;
  const size_t BHTD = (size_t)MHA_B * MHA_H * MHA_T * MHA_D;  // 8,388,608
  float* q  = ws;
  float* k  = ws + BHTD;
  float* v  = ws + 2 * BHTD;
  float* ao = ws + 3 * BHTD;  // (B, T, H*D)   total need: 4*BHTD*4B = 128 MB

  dim3 blk(256);
  mha_qkv_proj<<<dim3((MHA_B * MHA_T) / 128, MHA_H, 3), blk, 0, stream>>>(
      x, Wq, Wk, Wv, q, k, v);
  mha_flash_attn<<<dim3(MHA_T / 128, MHA_H, MHA_B), blk, 0, stream>>>(
      q, k, v, rel, ao);
  mha_out_proj<<<dim3((MHA_B * MHA_T) / 128, MHA_C / 64, 1), blk, 0, stream>>>(
      ao, Wp, bp, out);
}